// CNF_57054345560551
// MI455X (gfx1250) — compile-verified
//
#include <hip/hip_runtime.h>
#include <math.h>

typedef __attribute__((ext_vector_type(2))) float v2f;
typedef __attribute__((ext_vector_type(8))) float v8f;

#define B_SZ 1024
#define D_SZ 64
#define H_SZ 512
#define IN_SZ (D_SZ + 1)
#define NB 4   // column tiles per wave (16x64 output per wave)

__device__ __forceinline__ v8f wmma4(v2f a, v2f b, v8f c) {
  return __builtin_amdgcn_wmma_f32_16x16x4_f32(false, a, false, b,
                                               (short)0, c, false, false);
}

// ---------------------------------------------------------------------------
// Kernel 0: M[m,n] = W2[m,n] * sum_{j<64} W1[j,m] * W3[n,j]
// GEMM (512 x 64 x 512), A = W1[0:64,:]^T, B = W3^T, then Hadamard with W2.
// Each wave: 16 rows x 64 cols (4 independent WMMA chains).
// ---------------------------------------------------------------------------
__global__ __launch_bounds__(128) void k_prepM(const float* __restrict__ W1,
                                               const float* __restrict__ W2,
                                               const float* __restrict__ W3,
                                               float* __restrict__ M) {
  const unsigned waveId = blockIdx.x * (blockDim.x >> 5) + (threadIdx.x >> 5);
  const unsigned lane   = threadIdx.x & 31;
  const unsigned groups = H_SZ / (16 * NB);          // 8
  const unsigned tm  = waveId / groups;
  const unsigned tg  = waveId % groups;
  const unsigned mrow  = tm * 16 + (lane & 15);
  const unsigned col0  = tg * 16 * NB + (lane & 15);
  const unsigned kHalf = (lane >> 4) << 1;           // 0 or 2
  v8f acc[NB] = {};
  for (int k = 0; k < D_SZ; k += 4) {
    v2f a;
    a.x = W1[(k + kHalf)     * H_SZ + mrow];         // A[m,j] = W1[j,m]
    a.y = W1[(k + kHalf + 1) * H_SZ + mrow];
#pragma unroll
    for (int j = 0; j < NB; ++j) {
      v2f b;
      b.x = W3[(col0 + j * 16) * D_SZ + (k + kHalf)];     // B[j,n] = W3[n,j]
      b.y = W3[(col0 + j * 16) * D_SZ + (k + kHalf + 1)];
      acc[j] = wmma4(a, b, acc[j]);
    }
  }
  const unsigned rbase = tm * 16 + ((lane >> 4) << 3);
#pragma unroll
  for (int j = 0; j < NB; ++j) {
    const unsigned c = col0 + j * 16;
#pragma unroll
    for (int v = 0; v < 8; ++v) {
      const unsigned r = rbase + v;
      M[r * H_SZ + c] = W2[r * H_SZ + c] * acc[j][v];
    }
  }
}

// ---------------------------------------------------------------------------
// Kernel 1: z1 = x @ W1[0:64,:] + (t*W1[64,:] + b1); h1 = tanh(z1); a = 1-h1^2
// ---------------------------------------------------------------------------
__global__ __launch_bounds__(128) void k_layer1(const float* __restrict__ state,
                                                const float* __restrict__ tptr,
                                                const float* __restrict__ W1,
                                                const float* __restrict__ b1,
                                                float* __restrict__ H1,
                                                float* __restrict__ Aact) {
  const unsigned waveId = blockIdx.x * (blockDim.x >> 5) + (threadIdx.x >> 5);
  const unsigned lane   = threadIdx.x & 31;
  const unsigned groups = H_SZ / (16 * NB);          // 8
  const unsigned ti  = waveId / groups;              // batch tile 0..63
  const unsigned tg  = waveId % groups;
  const unsigned mrow  = ti * 16 + (lane & 15);
  const unsigned col0  = tg * 16 * NB + (lane & 15);
  const unsigned kHalf = (lane >> 4) << 1;
  const float t = tptr[0];
  v8f acc[NB] = {};
  for (int k = 0; k < D_SZ; k += 4) {
    v2f a;
    a.x = state[mrow * IN_SZ + 1 + k + kHalf];       // x = state[:,1:]
    a.y = state[mrow * IN_SZ + 2 + k + kHalf];
#pragma unroll
    for (int j = 0; j < NB; ++j) {
      v2f b;
      b.x = W1[(k + kHalf)     * H_SZ + col0 + j * 16];
      b.y = W1[(k + kHalf + 1) * H_SZ + col0 + j * 16];
      acc[j] = wmma4(a, b, acc[j]);
    }
  }
  const unsigned rbase = ti * 16 + ((lane >> 4) << 3);
#pragma unroll
  for (int j = 0; j < NB; ++j) {
    const unsigned c    = col0 + j * 16;
    const float   bias  = t * W1[D_SZ * H_SZ + c] + b1[c];
#pragma unroll
    for (int v = 0; v < 8; ++v) {
      const unsigned r = rbase + v;
      const float h = tanhf(acc[j][v] + bias);
      H1[r * H_SZ + c]   = h;
      Aact[r * H_SZ + c] = 1.0f - h * h;
    }
  }
}

// ---------------------------------------------------------------------------
// Kernel 2: z2 = h1 @ W2 + b2; h2 = tanh(z2); b = 1-h2^2
// ---------------------------------------------------------------------------
__global__ __launch_bounds__(128) void k_layer2(const float* __restrict__ H1,
                                                const float* __restrict__ W2,
                                                const float* __restrict__ b2,
                                                float* __restrict__ H2,
                                                float* __restrict__ Bact) {
  const unsigned waveId = blockIdx.x * (blockDim.x >> 5) + (threadIdx.x >> 5);
  const unsigned lane   = threadIdx.x & 31;
  const unsigned groups = H_SZ / (16 * NB);          // 8
  const unsigned ti  = waveId / groups;
  const unsigned tg  = waveId % groups;
  const unsigned mrow  = ti * 16 + (lane & 15);
  const unsigned col0  = tg * 16 * NB + (lane & 15);
  const unsigned kHalf = (lane >> 4) << 1;
  v8f acc[NB] = {};
  for (int k = 0; k < H_SZ; k += 4) {
    v2f a;
    a.x = H1[mrow * H_SZ + k + kHalf];
    a.y = H1[mrow * H_SZ + k + kHalf + 1];
#pragma unroll
    for (int j = 0; j < NB; ++j) {
      v2f b;
      b.x = W2[(k + kHalf)     * H_SZ + col0 + j * 16];
      b.y = W2[(k + kHalf + 1) * H_SZ + col0 + j * 16];
      acc[j] = wmma4(a, b, acc[j]);
    }
  }
  const unsigned rbase = ti * 16 + ((lane >> 4) << 3);
#pragma unroll
  for (int j = 0; j < NB; ++j) {
    const unsigned c   = col0 + j * 16;
    const float   bias = b2[c];
#pragma unroll
    for (int v = 0; v < 8; ++v) {
      const unsigned r = rbase + v;
      const float h = tanhf(acc[j][v] + bias);
      H2[r * H_SZ + c]   = h;
      Bact[r * H_SZ + c] = 1.0f - h * h;
    }
  }
}

// ---------------------------------------------------------------------------
// Kernel 3: dx = h2 @ W3 + b3  -> out[:, 1:65]
// N = 64 = exactly one NB-group per batch row tile.
// ---------------------------------------------------------------------------
__global__ __launch_bounds__(128) void k_layer3(const float* __restrict__ H2,
                                                const float* __restrict__ W3,
                                                const float* __restrict__ b3,
                                                float* __restrict__ out) {
  const unsigned waveId = blockIdx.x * (blockDim.x >> 5) + (threadIdx.x >> 5);
  const unsigned lane   = threadIdx.x & 31;
  const unsigned ti    = waveId;                     // batch tile 0..63
  const unsigned mrow  = ti * 16 + (lane & 15);
  const unsigned col0  = lane & 15;
  const unsigned kHalf = (lane >> 4) << 1;
  v8f acc[NB] = {};
  for (int k = 0; k < H_SZ; k += 4) {
    v2f a;
    a.x = H2[mrow * H_SZ + k + kHalf];
    a.y = H2[mrow * H_SZ + k + kHalf + 1];
#pragma unroll
    for (int j = 0; j < NB; ++j) {
      v2f b;
      b.x = W3[(k + kHalf)     * D_SZ + col0 + j * 16];
      b.y = W3[(k + kHalf + 1) * D_SZ + col0 + j * 16];
      acc[j] = wmma4(a, b, acc[j]);
    }
  }
  const unsigned rbase = ti * 16 + ((lane >> 4) << 3);
#pragma unroll
  for (int j = 0; j < NB; ++j) {
    const unsigned c   = col0 + j * 16;
    const float   bias = b3[c];
#pragma unroll
    for (int v = 0; v < 8; ++v) {
      const unsigned r = rbase + v;
      out[r * IN_SZ + 1 + c] = acc[j][v] + bias;
    }
  }
}

// ---------------------------------------------------------------------------
// Kernel 4: trJ_i = a_i^T M b_i ;  out[i,0] = -trJ_i
// One WG (4 waves) per 16-row batch tile; wave w handles NB-groups w, w+4
// (2 x 64 columns). V-tile = (A-tile @ M) via blocked WMMA, dotted with Bact,
// then LDS ds_add_f32 row reduction.
// ---------------------------------------------------------------------------
__global__ __launch_bounds__(128) void k_trace(const float* __restrict__ Aact,
                                               const float* __restrict__ M,
                                               const float* __restrict__ Bact,
                                               float* __restrict__ out) {
  __shared__ float sums[16];
  const unsigned wave = threadIdx.x >> 5;            // 0..3
  const unsigned lane = threadIdx.x & 31;
  const unsigned ti   = blockIdx.x;                  // batch tile 0..63
  if (threadIdx.x < 16) sums[threadIdx.x] = 0.0f;
  __syncthreads();

  const unsigned mrow  = ti * 16 + (lane & 15);
  const unsigned kHalf = (lane >> 4) << 1;
  const unsigned rbase = ti * 16 + ((lane >> 4) << 3);
  v8f tot = {};
  for (unsigned tg = wave; tg < H_SZ / (16 * NB); tg += 4) {  // 2 iterations
    const unsigned col0 = tg * 16 * NB + (lane & 15);
    v8f acc[NB] = {};
    for (int k = 0; k < H_SZ; k += 4) {
      v2f a;
      a.x = Aact[mrow * H_SZ + k + kHalf];
      a.y = Aact[mrow * H_SZ + k + kHalf + 1];
#pragma unroll
      for (int j = 0; j < NB; ++j) {
        v2f b;
        b.x = M[(k + kHalf)     * H_SZ + col0 + j * 16];
        b.y = M[(k + kHalf + 1) * H_SZ + col0 + j * 16];
        acc[j] = wmma4(a, b, acc[j]);
      }
    }
#pragma unroll
    for (int j = 0; j < NB; ++j) {
      const unsigned c = col0 + j * 16;
#pragma unroll
      for (int v = 0; v < 8; ++v) {
        const unsigned r = rbase + v;
        tot[v] += acc[j][v] * Bact[r * H_SZ + c];
      }
    }
  }
  const unsigned rloc = (lane >> 4) << 3;
#pragma unroll
  for (int v = 0; v < 8; ++v) atomicAdd(&sums[rloc + v], tot[v]);
  __syncthreads();
  if (threadIdx.x < 16)
    out[(ti * 16 + threadIdx.x) * IN_SZ] = -sums[threadIdx.x];
}

// ---------------------------------------------------------------------------
extern "C" void kernel_launch(void* const* d_in, const int* in_sizes, int n_in,
                              void* d_out, int out_size, void* d_ws, size_t ws_size,
                              hipStream_t stream) {
  const float* state = (const float*)d_in[0];   // (1024, 65)
  const float* t     = (const float*)d_in[1];   // (1,)
  const float* W1    = (const float*)d_in[2];   // (65, 512)
  const float* b1    = (const float*)d_in[3];   // (512,)
  const float* W2    = (const float*)d_in[4];   // (512, 512)
  const float* b2    = (const float*)d_in[5];   // (512,)
  const float* W3    = (const float*)d_in[6];   // (512, 64)
  const float* b3    = (const float*)d_in[7];   // (64,)
  float* out = (float*)d_out;                   // (1024, 65)

  float* ws   = (float*)d_ws;
  float* Mbuf = ws;                              // 512*512
  float* H1   = Mbuf + H_SZ * H_SZ;              // 1024*512
  float* Aact = H1   + B_SZ * H_SZ;              // 1024*512
  float* H2   = Aact + B_SZ * H_SZ;              // 1024*512
  float* Bact = H2   + B_SZ * H_SZ;              // 1024*512

  const dim3 blk(128);  // 4 wave32 per WG, each wave owns a 16x64 output slab
  // prepM:  32 row tiles * 8 groups = 256 waves -> 64 WGs
  k_prepM <<<64, blk, 0, stream>>>(W1, W2, W3, Mbuf);
  // layer1: 64 batch tiles * 8 groups = 512 waves -> 128 WGs
  k_layer1<<<128, blk, 0, stream>>>(state, t, W1, b1, H1, Aact);
  // layer2: 512 waves -> 128 WGs
  k_layer2<<<128, blk, 0, stream>>>(H1, W2, b2, H2, Bact);
  // layer3: 64 batch tiles * 1 group = 64 waves -> 16 WGs
  k_layer3<<<16, blk, 0, stream>>>(H2, W3, b3, out);
  // trace: one WG per 16-row batch tile
  k_trace <<<64, blk, 0, stream>>>(Aact, Mbuf, Bact, out);
}